// ColorConsistencyLoss_57982058496130
// MI455X (gfx1250) — compile-verified
//
#include <hip/hip_runtime.h>
#include <hip/hip_bf16.h>
#include <math.h>

#define NUM_LABELS 100
#define TILE_M 16
#define K_STEPS 25            // 25 * 4 = 100 labels exactly
#define LOG2E_F 1.44269504088896340736f
#define LN2_F   0.69314718055994530942f

typedef __attribute__((ext_vector_type(2))) float v2f;
typedef __attribute__((ext_vector_type(8))) float v8f;

// Fast exp(x) for x <= 0, |x| modest (no overflow/denorm fixup needed):
// single v_fma + v_exp_f32.
__device__ __forceinline__ float fast_exp_shift(float x, float neg_m_scaled) {
#if __has_builtin(__builtin_amdgcn_exp2f)
    return __builtin_amdgcn_exp2f(fmaf(x, LOG2E_F, neg_m_scaled));
#else
    return __expf(x + neg_m_scaled * LN2_F);
#endif
}

// Fast ln(s) for s in [1, NUM_LABELS]: v_log_f32 * ln2.
__device__ __forceinline__ float fast_log(float s) {
#if __has_builtin(__builtin_amdgcn_logf)
    return __builtin_amdgcn_logf(s) * LN2_F;
#else
    return __logf(s);
#endif
}

// ---------------------------------------------------------------------------
// Kernel 1: streaming cross-entropy. Each wave processes 16-row tiles.
// Row sums of exp(x - m) are computed with chained V_WMMA_F32_16X16X4_F32
// against an all-ones B matrix. Loads are 64-bit non-temporal (420MB stream,
// read-once, > 192MB L2).
// ---------------------------------------------------------------------------
__global__ void __launch_bounds__(256)
ce_rowloss_kernel(const float* __restrict__ logits,
                  const int* __restrict__ target,
                  float* __restrict__ nll_partials,
                  int HW)
{
    const int lane   = threadIdx.x & 31;
    const int waveIB = threadIdx.x >> 5;                 // wave in block (0..7)
    const int wavesPerBlock = blockDim.x >> 5;
    const int waveId = blockIdx.x * wavesPerBlock + waveIB;
    const int totalWaves = gridDim.x * wavesPerBlock;

    const int r = lane & 15;        // row within tile (M)
    const int g = lane >> 4;        // K-group: 0 -> K{0,1}, 1 -> K{2,3}
    const int numTiles = HW / TILE_M;

    // B = all ones: every element of every B VGPR is 1.0 regardless of layout.
    v2f bOnes; bOnes.x = 1.0f; bOnes.y = 1.0f;

    float acc = 0.0f;               // per-lane NLL accumulator

    for (int tile = waveId; tile < numTiles; tile += totalWaves) {
        const int row = tile * TILE_M + r;
        const float* p = logits + (size_t)row * NUM_LABELS + 2 * g;

        // Cache this lane's 50 row elements (25 x float2), track local max.
        // Per ISA A-layout (16x4 f32): for WMMA step t this lane supplies
        // columns 4t + 2g and 4t + 2g + 1.
        v2f v[K_STEPS];
        float m = -INFINITY;
        #pragma unroll
        for (int t = 0; t < K_STEPS; ++t) {
            v[t] = __builtin_nontemporal_load((const v2f*)(p + 4 * t));
            m = fmaxf(m, fmaxf(v[t].x, v[t].y));
        }
        // Combine the two K-groups of the same row -> full row max in both.
        m = fmaxf(m, __shfl_xor(m, 16));
        const float neg_m_scaled = -m * LOG2E_F;

        // Sum exp(x - m) over K=100 via 25 chained WMMAs (D = A*ones + C).
        // exp lowers to v_fma + v_exp_f32 (arg <= 0, no fixup needed).
        v8f d = {};
        #pragma unroll
        for (int t = 0; t < K_STEPS; ++t) {
            v2f a;
            a.x = fast_exp_shift(v[t].x, neg_m_scaled);
            a.y = fast_exp_shift(v[t].y, neg_m_scaled);
            d = __builtin_amdgcn_wmma_f32_16x16x4_f32(
                    false, a, false, bOnes, (short)0, d, false, false);
        }

        // D layout: VGPR dIdx, lane 0 holds S_dIdx, lane 16 holds S_{dIdx+8}.
        // Scatter so lane r ends with S_r.
        float S = 0.0f;
        #pragma unroll
        for (int dIdx = 0; dIdx < 8; ++dIdx) {
            float lo = __shfl(d[dIdx], 0);
            float hi = __shfl(d[dIdx], 16);
            if (r == dIdx)     S = lo;
            if (r == dIdx + 8) S = hi;
        }

        // One K-group computes the NLL for its row: logsumexp - x_target.
        if (g == 0) {
            int   t  = target[row];
            float xt = logits[(size_t)row * NUM_LABELS + t];
            acc += (fast_log(S) + m) - xt;
        }
    }

    // Remainder rows (HW not multiple of 16): scalar fallback, first wave only.
    const int rem = HW - numTiles * TILE_M;
    if (rem > 0 && blockIdx.x == 0 && waveIB == 0 && lane < rem) {
        const int row = numTiles * TILE_M + lane;
        const float* p = logits + (size_t)row * NUM_LABELS;
        float m = -INFINITY;
        for (int k = 0; k < NUM_LABELS; ++k) m = fmaxf(m, p[k]);
        const float nms = -m * LOG2E_F;
        float s = 0.0f;
        for (int k = 0; k < NUM_LABELS; ++k) s += fast_exp_shift(p[k], nms);
        acc += (fast_log(s) + m) - p[target[row]];
    }

    // Deterministic block reduction: wave shuffle tree, then in-order LDS sum.
    #pragma unroll
    for (int off = 16; off > 0; off >>= 1) acc += __shfl_xor(acc, off);

    __shared__ float wpart[8];
    if (lane == 0) wpart[waveIB] = acc;
    __syncthreads();
    if (threadIdx.x == 0) {
        float s = 0.0f;
        for (int i = 0; i < wavesPerBlock; ++i) s += wpart[i];
        nll_partials[blockIdx.x] = s;
    }
}

// ---------------------------------------------------------------------------
// Kernel 2: per-block privatized segment sums of img and img^2 over labels.
// LDS bins (100 labels x 3 channels x {sum, sumsq}) via ds_add_f32 atomics;
// each block writes its 600-float partial to workspace (no global atomics).
// ---------------------------------------------------------------------------
__global__ void __launch_bounds__(256)
segsum_kernel(const float* __restrict__ img,
              const int* __restrict__ target,
              float* __restrict__ bins_out,   // [gridDim.x][600]
              int HW)
{
    __shared__ float s[600];
    for (int i = threadIdx.x; i < 600; i += blockDim.x) s[i] = 0.0f;
    __syncthreads();

    for (int p = blockIdx.x * blockDim.x + threadIdx.x; p < HW;
         p += gridDim.x * blockDim.x) {
        const int lab = target[p];
        const float a = img[3 * (size_t)p + 0];
        const float b = img[3 * (size_t)p + 1];
        const float c = img[3 * (size_t)p + 2];
        atomicAdd(&s[lab * 3 + 0], a);
        atomicAdd(&s[lab * 3 + 1], b);
        atomicAdd(&s[lab * 3 + 2], c);
        atomicAdd(&s[300 + lab * 3 + 0], a * a);
        atomicAdd(&s[300 + lab * 3 + 1], b * b);
        atomicAdd(&s[300 + lab * 3 + 2], c * c);
    }
    __syncthreads();

    float* out = bins_out + (size_t)blockIdx.x * 600;
    for (int i = threadIdx.x; i < 600; i += blockDim.x) out[i] = s[i];
}

// ---------------------------------------------------------------------------
// Kernel 3: deterministic finalize. Fixed-order sums of partials, per-bin
// std, sigmoid, blend.
// ---------------------------------------------------------------------------
__global__ void __launch_bounds__(512)
finalize_kernel(const float* __restrict__ nll_partials, int nb1,
                const float* __restrict__ bins, int nb2,
                int HW, float* __restrict__ out)
{
    __shared__ float red[512];
    __shared__ float loss1_s;
    const int tid = threadIdx.x;

    // ---- loss 1: mean NLL ----
    float s = 0.0f;
    for (int i = tid; i < nb1; i += 512) s += nll_partials[i];
    red[tid] = s;
    __syncthreads();
    for (int off = 256; off > 0; off >>= 1) {
        if (tid < off) red[tid] += red[tid + off];
        __syncthreads();
    }
    if (tid == 0) loss1_s = red[0] / (float)HW;
    __syncthreads();

    // ---- loss 2: sum of per-(label,channel) population std ----
    float sv = 0.0f;
    if (tid < 300) {
        float s1 = 0.0f, s2 = 0.0f;
        for (int b = 0; b < nb2; ++b) {
            s1 += bins[(size_t)b * 600 + tid];
            s2 += bins[(size_t)b * 600 + 300 + tid];
        }
        const float invHW = 1.0f / (float)HW;
        const float mean  = s1 * invHW;
        const float var   = fmaxf(s2 * invHW - mean * mean, 0.0f);
        sv = sqrtf(var);
    }
    red[tid] = sv;
    __syncthreads();
    for (int off = 256; off > 0; off >>= 1) {
        if (tid < off) red[tid] += red[tid + off];
        __syncthreads();
    }
    if (tid == 0) {
        const float std_all = red[0];
        const float loss2 = 1.0f / (1.0f + expf(-std_all));
        out[0] = 0.7f * loss1_s + 0.3f * loss2;
    }
}

// ---------------------------------------------------------------------------
extern "C" void kernel_launch(void* const* d_in, const int* in_sizes, int n_in,
                              void* d_out, int out_size, void* d_ws, size_t ws_size,
                              hipStream_t stream)
{
    const float* logits = (const float*)d_in[0];   // [HW, 100]
    const int*   target = (const int*)d_in[1];     // [HW]
    const float* img    = (const float*)d_in[2];   // [H, W, 3] flat
    float*       out    = (float*)d_out;

    const int HW = in_sizes[1];

    const int NB1 = 512;   // blocks for CE kernel (4096 waves, 16 tiles each)
    const int NB2 = 128;   // blocks for segment-sum kernel

    float* ws_nll  = (float*)d_ws;       // NB1 floats
    float* ws_bins = ws_nll + NB1;       // NB2 * 600 floats (~307 KB total)

    ce_rowloss_kernel<<<NB1, 256, 0, stream>>>(logits, target, ws_nll, HW);
    segsum_kernel<<<NB2, 256, 0, stream>>>(img, target, ws_bins, HW);
    finalize_kernel<<<1, 512, 0, stream>>>(ws_nll, NB1, ws_bins, NB2, HW, out);
}